// MRSTFTLoss_Mel_40862318854564
// MI455X (gfx1250) — compile-verified
//
#include <hip/hip_runtime.h>
#include <math.h>

typedef __attribute__((ext_vector_type(2))) float v2f;
typedef __attribute__((ext_vector_type(8))) float v8f;

#define MU      (-9.1232f)
#define SIGMA   (3.5412f)
#define EPS_MAG (1e-7f)
#define EPS_SC  (1e-8f)
#define LN2     (0.69314718055994531f)
#define TLEN    1024
#define NROWS   4096

// One workgroup = one row at resolution RES. 256 threads = 8 waves (wave32).
// All strides/bounds are compile-time constants -> immediate DS offsets,
// fully unrolled K loop, minimal address arithmetic around the WMMAs.
template <int RES>
__global__ __launch_bounds__(256) void mrstft_partials(const float* __restrict__ x,
                                                       const float* __restrict__ y,
                                                       float* __restrict__ partial) {
  constexpr int N_FFT   = 8 << RES;                  // 8,16,32,64
  constexpr int HOP     = 2 << RES;                  // 2,4,8,16
  constexpr int LHOP    = 1 + RES;                   // log2(HOP)
  constexpr int NB      = (N_FFT >> 1) + 1;          // onesided bins
  constexpr int PAD     = N_FFT >> 1;
  constexpr int NFRAMES = 1 + TLEN / HOP;            // 513,257,129,65
  constexpr int PADN    = ((2 * NB + 15) / 16) * 16; // re/im interleaved, 16-pad
  constexpr int PLEN    = TLEN + N_FFT;
  constexpr int FT      = (NFRAMES + 15) / 16;       // frame tiles
  constexpr int NT      = PADN / 16;                 // column tiles
  constexpr int TOTAL   = FT * NT;

  const int row = blockIdx.x;

  __shared__ float sx[PLEN];            // padded, un-normalized x row
  __shared__ float sy[PLEN];            // padded, un-normalized y row
  __shared__ float sb[N_FFT * PADN];    // windowed DFT basis, K x PADN
  __shared__ float red[8 * 3];          // per-wave partials

  const int tid = threadIdx.x;

  // ---- stage reflect-padded exp(v*SIGMA+MU) signals ----
  for (int p = tid; p < PLEN; p += 256) {
    int s = p - PAD;
    s = (s < 0) ? -s : s;
    s = (s >= TLEN) ? (2 * TLEN - 2 - s) : s;
    sx[p] = __expf(x[(size_t)row * TLEN + s] * SIGMA + MU);
    sy[p] = __expf(y[(size_t)row * TLEN + s] * SIGMA + MU);
  }

  // ---- stage windowed DFT basis: col 2b = win*cos, col 2b+1 = -win*sin ----
  constexpr float TWOPI_N = 6.28318530717958647f / (float)N_FFT;
  for (int idx = tid; idx < N_FFT * PADN; idx += 256) {
    int k = idx / PADN;
    int j = idx - k * PADN;
    int b = j >> 1;
    float v = 0.0f;
    if (b < NB) {
      float win = 0.5f * (1.0f - __cosf(TWOPI_N * (float)k)); // periodic Hann
      int kb = (k * b) & (N_FFT - 1);                         // arg reduction
      float sn, cs;
      __sincosf(TWOPI_N * (float)kb, &sn, &cs);
      v = (j & 1) ? (-win * sn) : (win * cs);
    }
    sb[idx] = v;
  }
  __syncthreads();

  const int lane  = tid & 31;
  const int wave  = tid >> 5;
  const int laneM = lane & 15;  // M / N position within tile
  const int laneH = lane >> 4;  // half-wave select

  float acc_d2 = 0.f, acc_y2 = 0.f, acc_l1 = 0.f;

  for (int t = wave; t < TOTAL; t += 8) {  // wave-uniform -> EXEC all-ones
    int ft = t / NT;                       // constant divisor
    int nt = t - ft * NT;
    int m0 = ft << 4;
    int n0 = nt << 4;

    v8f cx = {};
    v8f cy = {};

    int mA = m0 + laneM;
    int abase = (mA < NFRAMES) ? (mA << LHOP) : 0;  // clamp ghost frames
    int kA0 = laneH << 1;                           // K pair by half-wave
    const float* ax_p = &sx[abase + kA0];
    const float* ay_p = &sy[abase + kA0];
    const float* bb_p = &sb[kA0 * PADN + n0 + laneM];

#pragma unroll
    for (int k0 = 0; k0 < N_FFT; k0 += 4) {
      v2f ax, ay, bb;
      ax[0] = ax_p[k0];
      ax[1] = ax_p[k0 + 1];
      ay[0] = ay_p[k0];
      ay[1] = ay_p[k0 + 1];
      bb[0] = bb_p[k0 * PADN];
      bb[1] = bb_p[k0 * PADN + PADN];
      cx = __builtin_amdgcn_wmma_f32_16x16x4_f32(false, ax, false, bb,
                                                 (short)0, cx, false, false);
      cy = __builtin_amdgcn_wmma_f32_16x16x4_f32(false, ay, false, bb,
                                                 (short)0, cy, false, false);
    }

    // C layout: VGPR r holds (M = r + 8*laneH, N = laneM). Even column = real,
    // odd = imag of same bin; partner lane = lane^1 (stays within the half).
    // Branchless: every intermediate is finite (mag >= EPS_MAG), so mask the
    // addends with selects instead of branching. Log-L1 is accumulated in the
    // log2 domain (raw v_log_f32) and scaled by ln2 once at the end.
    int j = n0 + laneM;
    bool isre = ((j & 1) == 0) && (j < 2 * NB);
#pragma unroll
    for (int r = 0; r < 8; ++r) {
      float rex = cx[r];
      float imx = __shfl_xor(rex, 1, 32);
      float rey = cy[r];
      float imy = __shfl_xor(rey, 1, 32);
      int frame = m0 + r + (laneH << 3);
      bool valid = isre && (frame < NFRAMES);
      float xm = __builtin_amdgcn_sqrtf(rex * rex + imx * imx) + EPS_MAG;
      float ym = __builtin_amdgcn_sqrtf(rey * rey + imy * imy) + EPS_MAG;
      float d = ym - xm;
      float l2 = __builtin_amdgcn_logf(xm) - __builtin_amdgcn_logf(ym);
      acc_d2 += valid ? (d * d) : 0.0f;
      acc_y2 += valid ? (ym * ym) : 0.0f;
      acc_l1 += valid ? fabsf(l2) : 0.0f;
    }
  }

  // ---- wave-level pre-reduction, then cross-wave via tiny LDS ----
#pragma unroll
  for (int s = 16; s > 0; s >>= 1) {
    acc_d2 += __shfl_down(acc_d2, s, 32);
    acc_y2 += __shfl_down(acc_y2, s, 32);
    acc_l1 += __shfl_down(acc_l1, s, 32);
  }
  if (lane == 0) {
    red[wave * 3 + 0] = acc_d2;
    red[wave * 3 + 1] = acc_y2;
    red[wave * 3 + 2] = acc_l1;
  }
  __syncthreads();
  if (tid == 0) {
    float d2 = 0.f, y2 = 0.f, l1 = 0.f;
#pragma unroll
    for (int w = 0; w < 8; ++w) {
      d2 += red[w * 3 + 0];
      y2 += red[w * 3 + 1];
      l1 += red[w * 3 + 2];
    }
    float* o = partial + ((size_t)RES * NROWS + row) * 3;
    o[0] = d2;
    o[1] = y2;
    o[2] = l1 * LN2;  // back from log2 to natural-log domain
  }
}

// Deterministic fixed-order final reduction (no float atomics).
__global__ __launch_bounds__(256) void mrstft_reduce(const float* __restrict__ partial,
                                                     float* __restrict__ out) {
  __shared__ float red[256];
  const int tid = threadIdx.x;
  float loss_sc = 0.f, loss_mag = 0.f;

  for (int res = 0; res < 4; ++res) {
    float comp[3];
    for (int c = 0; c < 3; ++c) {
      float s = 0.f;
      for (int i = tid; i < NROWS; i += 256)
        s += partial[((size_t)res * NROWS + i) * 3 + c];
      red[tid] = s;
      __syncthreads();
      for (int k = 128; k > 0; k >>= 1) {
        if (tid < k) red[tid] += red[tid + k];
        __syncthreads();
      }
      comp[c] = red[0];
      __syncthreads();
    }
    int hop = 2 << res;
    int nb = (8 << res) / 2 + 1;
    float nframes = (float)(1 + TLEN / hop);
    float count = (float)NROWS * nframes * (float)nb;
    loss_sc  += sqrtf(comp[0]) / (sqrtf(comp[1]) + EPS_SC);
    loss_mag += comp[2] / count;
  }
  if (tid == 0) {
    out[0] = loss_sc * 0.25f;
    out[1] = loss_mag * 0.25f;
  }
}

extern "C" void kernel_launch(void* const* d_in, const int* in_sizes, int n_in,
                              void* d_out, int out_size, void* d_ws, size_t ws_size,
                              hipStream_t stream) {
  const float* x = (const float*)d_in[0];
  const float* y = (const float*)d_in[1];
  float* partial = (float*)d_ws;  // 4 * 4096 * 3 floats = 192 KB

  mrstft_partials<0><<<dim3(NROWS), 256, 0, stream>>>(x, y, partial);
  mrstft_partials<1><<<dim3(NROWS), 256, 0, stream>>>(x, y, partial);
  mrstft_partials<2><<<dim3(NROWS), 256, 0, stream>>>(x, y, partial);
  mrstft_partials<3><<<dim3(NROWS), 256, 0, stream>>>(x, y, partial);
  mrstft_reduce<<<1, 256, 0, stream>>>(partial, (float*)d_out);
}